// CRFLayer_50148038148245
// MI455X (gfx1250) — compile-verified
//
#include <hip/hip_runtime.h>
#include <stdint.h>

// CRF Viterbi decode, B sequences x T steps x K=64 tags.
// (max,+) semiring => WMMA not applicable; kernel is VALU-bound.
// Hot loop hand-packed with V_PK_ADD_F32 (VOP3P, dword op_sel broadcasts) + v_max3_f32.
// Forward stores out_t vectors (f32); backtrace recomputes the needed bp column.
// feats stream staged into LDS with the CDNA5 Tensor Data Mover (TDM).

#define KK 64
#define START_TAG_ 1
#define END_TAG_ 2
#define NEGV -10000.0f
#define NWAVES 8   // waves per block (one sequence per wave)
#define TS 8       // feats rows per TDM chunk (double buffered)

#if defined(__HIP_DEVICE_COMPILE__) && \
    __has_builtin(__builtin_amdgcn_tensor_load_to_lds) && \
    __has_builtin(__builtin_amdgcn_s_wait_tensorcnt)
#define HAVE_TDM 1
#else
#define HAVE_TDM 0
#endif

#if defined(__HIP_DEVICE_COMPILE__) && defined(__gfx1250__)
#define HAVE_PK 1
#else
#define HAVE_PK 0
#endif

typedef float v2f __attribute__((ext_vector_type(2)));
typedef unsigned int v4u_t __attribute__((ext_vector_type(4)));
typedef int v8i_t __attribute__((ext_vector_type(8)));
typedef int v4i_t __attribute__((ext_vector_type(4)));

__device__ __forceinline__ void wbar() {
#if defined(__HIP_DEVICE_COMPILE__)
  __builtin_amdgcn_wave_barrier();
#endif
}

__device__ __forceinline__ float fmax3f(float a, float b, float c) {
  return __builtin_fmaxf(__builtin_fmaxf(a, b), c);
}

// d = (a.x + b.x, a.y + b.y)
__device__ __forceinline__ v2f pk_add(v2f a, v2f b) {
#if HAVE_PK
  v2f d;
  asm("v_pk_add_f32 %0, %1, %2" : "=v"(d) : "v"(a), "v"(b));
  return d;
#else
  v2f d; d.x = a.x + b.x; d.y = a.y + b.y; return d;
#endif
}

// d = (a.x + b.x, a.y + b.x)  -- broadcast low dword of b to both sub-ops
__device__ __forceinline__ v2f pk_add_blo(v2f a, v2f b) {
#if HAVE_PK
  v2f d;
  asm("v_pk_add_f32 %0, %1, %2 op_sel:[0,0] op_sel_hi:[1,0]"
      : "=v"(d) : "v"(a), "v"(b));
  return d;
#else
  v2f d; d.x = a.x + b.x; d.y = a.y + b.x; return d;
#endif
}

// d = (a.x + b.y, a.y + b.y)  -- broadcast high dword of b to both sub-ops
__device__ __forceinline__ v2f pk_add_bhi(v2f a, v2f b) {
#if HAVE_PK
  v2f d;
  asm("v_pk_add_f32 %0, %1, %2 op_sel:[0,1] op_sel_hi:[1,1]"
      : "=v"(d) : "v"(a), "v"(b));
  return d;
#else
  v2f d; d.x = a.x + b.y; d.y = a.y + b.y; return d;
#endif
}

#if HAVE_TDM
// Build a 2D D# (tile = nrows x 64 f32, row stride 64) and issue TENSOR_LOAD_TO_LDS.
__device__ __forceinline__ void tdm_load_rows(const float* gsrc, const void* ldsDst,
                                              int nrows) {
  const uint64_t ga = (uint64_t)(uintptr_t)gsrc;
  const uint32_t lo = (uint32_t)(uintptr_t)ldsDst;  // flat addr[31:0] == LDS byte offset
  v4u_t g0;
  g0.x = 1u;                                         // count=1, user mode
  g0.y = lo;                                         // lds_addr
  g0.z = (uint32_t)ga;                               // global_addr[31:0]
  g0.w = ((uint32_t)(ga >> 32) & 0x01FFFFFFu) | (2u << 30);  // addr[56:32], type=2
  v8i_t g1 = (v8i_t)(0);
  g1.s0 = (int)(2u << 16);                 // data_size = 2 (4 bytes)
  g1.s1 = (int)((uint32_t)KK << 16);       // tensor_dim0 = 64
  g1.s2 = (int)((uint32_t)nrows << 16);    // tensor_dim1 = nrows
  g1.s3 = (int)((uint32_t)KK << 16);       // tile_dim0 = 64
  g1.s4 = nrows;                           // tile_dim1 = nrows
  g1.s5 = KK;                              // tensor_dim0_stride = 64
  const v4i_t gz = (v4i_t)(0);
#if __clang_major__ >= 23
  __builtin_amdgcn_tensor_load_to_lds(g0, g1, gz, gz, (v8i_t)(0), 0);
#else
  __builtin_amdgcn_tensor_load_to_lds(g0, g1, gz, gz, 0);
#endif
}
#endif

// One Viterbi step for this lane's two columns (j0, j0+1), packed f32 math.
// Mirrors reference rounding exactly: ((trans + feats) [+ end_adj]) + out, max over i.
template <bool LAST>
__device__ __forceinline__ void reduce_step(const float* outv, const float* transP,
                                            v2f fv, v2f adjv, int j0,
                                            float& na, float& nb) {
  float ma = -__builtin_inff(), mb = -__builtin_inff();
#pragma unroll
  for (int i2 = 0; i2 < KK / 2; ++i2) {
    const v2f o = *(const v2f*)(outv + 2 * i2);                             // broadcast pair
    v2f lt0 = pk_add(*(const v2f*)(transP + (size_t)(2 * i2) * KK + j0), fv);
    v2f lt1 = pk_add(*(const v2f*)(transP + (size_t)(2 * i2 + 1) * KK + j0), fv);
    if (LAST) { lt0 = pk_add(lt0, adjv); lt1 = pk_add(lt1, adjv); }
    const v2f c0 = pk_add_blo(lt0, o);     // + out[2*i2]   (row 2*i2, both cols)
    const v2f c1 = pk_add_bhi(lt1, o);     // + out[2*i2+1] (row 2*i2+1, both cols)
    ma = fmax3f(ma, c0.x, c1.x);           // v_max3_f32, column j0
    mb = fmax3f(mb, c0.y, c1.y);           // v_max3_f32, column j1
  }
  na = ma;
  nb = mb;
}

__global__ __launch_bounds__(NWAVES * 32)
void crf_viterbi_kernel(const float* __restrict__ feats, const int* __restrict__ leng,
                        const float* __restrict__ trans, float* __restrict__ outp,
                        float* __restrict__ hist, int B, int T) {
  __shared__ float transP[KK * KK];       // trans[i][j], row-major (conflict-free b64 reads)
  __shared__ float outv_s[NWAVES][KK];    // per-wave Viterbi score vector
#if HAVE_TDM
  __shared__ float fbuf[NWAVES][2][TS][KK];  // per-wave feats double buffer (TDM dest)
#endif
  const int tid = (int)threadIdx.x;
  const int wave = tid >> 5;
  const int lane = tid & 31;
  for (int k = tid; k < KK * KK; k += (int)blockDim.x) transP[k] = trans[k];
  __syncthreads();

  const int b = (int)blockIdx.x * NWAVES + wave;
  if (b >= B) return;
  const int L = leng[b];                  // 1 <= L <= T
  const int j0 = lane * 2, j1 = j0 + 1;
  float* outv = outv_s[wave];

  // init: START=0, else NEG
  v2f iv;
  iv.x = (j0 == START_TAG_) ? 0.0f : NEGV;
  iv.y = (j1 == START_TAG_) ? 0.0f : NEGV;
  *(v2f*)(outv + j0) = iv;
  wbar();
  const size_t baseF = (size_t)b * (size_t)T * KK;
  float* histB = hist + baseF;
  *(v2f*)(histB + j0) = iv;               // out_hist[0] = init

  v2f adjv;                               // end_adj columns
  adjv.x = (j0 == END_TAG_) ? 0.0f : NEGV;
  adjv.y = (j1 == END_TAG_) ? 0.0f : NEGV;

#if HAVE_TDM
  int buf = 0;
  if (L > 1) {
    int n0 = (L - 1 < TS) ? (L - 1) : TS;
    tdm_load_rows(feats + baseF + KK, &fbuf[wave][0][0][0], n0);
  }
#endif

  // ---------------- forward ----------------
  for (int t = 1; t < L; ++t) {
#if HAVE_TDM
    const int local = (t - 1) & (TS - 1);
    if (local == 0) {
      __builtin_amdgcn_s_wait_tensorcnt(0);     // current buffer filled
      asm volatile("" ::: "memory");
      const int nstart = t + TS;                // prefetch next chunk into other buffer
      if (nstart < L) {
        int nn = L - nstart; if (nn > TS) nn = TS;
        tdm_load_rows(feats + baseF + (size_t)nstart * KK, &fbuf[wave][buf ^ 1][0][0], nn);
      }
    }
    const v2f f = *(const v2f*)(&fbuf[wave][buf][local][j0]);
#else
    const v2f f = *(const v2f*)(feats + baseF + (size_t)t * KK + j0);
#endif
    float na, nb;
    if (t == L - 1) reduce_step<true >(outv, transP, f, adjv, j0, na, nb);
    else            reduce_step<false>(outv, transP, f, adjv, j0, na, nb);
    wbar();
    v2f nv; nv.x = na; nv.y = nb;
    *(v2f*)(outv + j0) = nv;                            // update out (LDS)
    *(v2f*)(histB + (size_t)t * KK + j0) = nv;          // out_hist[t]
    wbar();
#if HAVE_TDM
    if (local == TS - 1) buf ^= 1;
#endif
  }

  // ---------------- end tag: argmax over out_end (first-occurrence ties) ---------
  wbar();
  const v2f fe = *(const v2f*)(outv + j0);
  float v = fe.x;
  int idx = j0;
  if (fe.y > v) { v = fe.y; idx = j1; }
#pragma unroll
  for (int m = 16; m >= 1; m >>= 1) {
    const float v2 = __shfl_xor(v, m, 32);
    const int i2 = __shfl_xor(idx, m, 32);
    if (v2 > v || (v2 == v && i2 < idx)) { v = v2; idx = i2; }
  }
  if (lane == 0) {
    outp[(size_t)B * T + b] = v;                     // tag_scores[b]
    outp[(size_t)b * T + (L - 1)] = (float)idx;      // tag at last valid step
  }
  for (int tt = L + lane; tt < T; tt += 32)          // zero padding region
    outp[(size_t)b * T + tt] = 0.0f;

  // ---------------- backtrace: recompute bp[t+1][cur] from out_hist[t] ----------
  int cur = idx;
  for (int t = L - 2; t >= 0; --t) {
    if (t > 0) __builtin_prefetch(histB + (size_t)(t - 1) * KK, 0, 1);
    const float fn = feats[baseF + (size_t)(t + 1) * KK + cur];
    const float aj = ((t + 1 == L - 1) && (cur != END_TAG_)) ? NEGV : 0.0f;
    const v2f oo = *(const v2f*)(histB + (size_t)t * KK + j0);
    const float ta = trans[(size_t)j0 * KK + cur];   // trans[i][cur], i = j0/j1
    const float tb = trans[(size_t)j1 * KK + cur];
    float cv = ((ta + fn) + aj) + oo.x;              // reference association order
    int ci = j0;
    const float cb = ((tb + fn) + aj) + oo.y;
    if (cb > cv) { cv = cb; ci = j1; }
#pragma unroll
    for (int m = 16; m >= 1; m >>= 1) {
      const float v2 = __shfl_xor(cv, m, 32);
      const int i2 = __shfl_xor(ci, m, 32);
      if (v2 > cv || (v2 == cv && i2 < ci)) { cv = v2; ci = i2; }
    }
    cur = ci;
    if (lane == 0) outp[(size_t)b * T + t] = (float)cur;
  }
}

extern "C" void kernel_launch(void* const* d_in, const int* in_sizes, int n_in,
                              void* d_out, int out_size, void* d_ws, size_t ws_size,
                              hipStream_t stream) {
  (void)n_in; (void)out_size; (void)ws_size;
  const float* feats = (const float*)d_in[0];
  const int* leng = (const int*)d_in[1];
  const float* trans = (const float*)d_in[2];
  float* outp = (float*)d_out;
  float* hist = (float*)d_ws;  // needs B*T*64*4 bytes (256 MB for B=2048, T=512)
  const int B = in_sizes[1];
  const int T = in_sizes[0] / (B * KK);
  const int blocks = (B + NWAVES - 1) / NWAVES;
  crf_viterbi_kernel<<<blocks, NWAVES * 32, 0, stream>>>(feats, leng, trans, outp, hist, B, T);
}